// CELossWithLS_39968965656966
// MI455X (gfx1250) — compile-verified
//
#include <hip/hip_runtime.h>

#define CLASSES    10000
#define VEC4       (CLASSES / 4)        // 2500 float4 chunks per row
#define SMOOTH     0.1f
#define COMPL      0.9f
#define IGN        (-1)
#define NTHREADS   256
#define NWAVES     (NTHREADS / 32)      // wave32 on gfx1250

typedef __attribute__((ext_vector_type(2))) float v2f;
typedef __attribute__((ext_vector_type(8))) float v8f;

// Exact parameter types for the async global->LDS builtins (from hipcc diag:
// "int __vector(4) *"), with address-space qualification.
typedef int v4i __attribute__((vector_size(16)));
typedef __attribute__((address_space(1))) v4i* gbl_v4i_p;
typedef __attribute__((address_space(3))) v4i* lds_v4i_p;

#if __has_builtin(__builtin_amdgcn_global_load_async_to_lds_b128)
#define HAVE_ASYNC_LDS 1
#else
#define HAVE_ASYNC_LDS 0
#endif

// ---------- wave32 reductions ----------

// Full 32-lane f32 sum using V_WMMA_F32_16X16X4_F32.
// A layout (16x4 f32): lanes 0-15 -> VGPR0=K0, VGPR1=K1; lanes 16-31 -> VGPR0=K2, VGPR1=K3.
// With a={v,0}: A[m][0]=v_m, A[m][2]=v_{m+16}. B=ones => D[m][n] = v_m + v_{m+16}.
// D layout: lane n<16 holds M=0..7 in c[0..7]; lane n>=16 holds M=8..15.
// Sum of the 8 accumulators gives half-sums; one xor-16 shuffle completes the total.
__device__ __forceinline__ float wave_sum(float v) {
#if __has_builtin(__builtin_amdgcn_wmma_f32_16x16x4_f32)
  v2f a; a[0] = v;    a[1] = 0.0f;
  v2f b; b[0] = 1.0f; b[1] = 1.0f;
  v8f c = {};
  c = __builtin_amdgcn_wmma_f32_16x16x4_f32(
      /*neg_a=*/false, a, /*neg_b=*/false, b,
      /*c_mod=*/(short)0, c, /*reuse_a=*/false, /*reuse_b=*/false);
  float r = ((c[0] + c[1]) + (c[2] + c[3])) + ((c[4] + c[5]) + (c[6] + c[7]));
  r += __shfl_xor(r, 16, 32);
  return r;
#else
  #pragma unroll
  for (int off = 16; off > 0; off >>= 1) v += __shfl_xor(v, off, 32);
  return v;
#endif
}

__device__ __forceinline__ float wave_max(float v) {
  #pragma unroll
  for (int off = 16; off > 0; off >>= 1) v = fmaxf(v, __shfl_xor(v, off, 32));
  return v;
}

__device__ __forceinline__ float block_sum(float v, float* red, int tid) {
  float w = wave_sum(v);
  __syncthreads();                       // guard red[] reuse across calls
  if ((tid & 31) == 0) red[tid >> 5] = w;
  __syncthreads();
  float s = red[0];
  #pragma unroll
  for (int i = 1; i < NWAVES; ++i) s += red[i];
  return s;
}

__device__ __forceinline__ float block_max(float v, float* red, int tid) {
  float w = wave_max(v);
  __syncthreads();
  if ((tid & 31) == 0) red[tid >> 5] = w;
  __syncthreads();
  float m = red[0];
  #pragma unroll
  for (int i = 1; i < NWAVES; ++i) m = fmaxf(m, red[i]);
  return m;
}

// ---------- main kernel: one workgroup per token ----------

__global__ __launch_bounds__(NTHREADS) void focal_ls_ce_rows(
    const float* __restrict__ logits, const int* __restrict__ target,
    float* __restrict__ tok_loss, float* __restrict__ tok_mask) {
  __shared__ __align__(16) float row[CLASSES];
  __shared__ float red[NWAVES];

  const int tok = blockIdx.x;
  const int tid = threadIdx.x;
  const int tgt = target[tok];           // uniform scalar load

  if (tgt == IGN) {                      // uniform early-out: EXEC stays all-ones below
    if (tid == 0) { tok_loss[tok] = 0.0f; tok_mask[tok] = 0.0f; }
    return;
  }

  const float4* __restrict__ src = (const float4*)(logits + (size_t)tok * CLASSES);
  float4* dstv = (float4*)row;

  // ---- stage 40KB row into LDS (async gfx1250 path if available) ----
#if HAVE_ASYNC_LDS
  for (int i = tid; i < VEC4; i += NTHREADS) {
    __builtin_amdgcn_global_load_async_to_lds_b128(
        (gbl_v4i_p)(src + i),
        (lds_v4i_p)(dstv + i),
        0, 0);
  }
#if __has_builtin(__builtin_amdgcn_s_wait_asynccnt)
  __builtin_amdgcn_s_wait_asynccnt(0);
#else
  asm volatile("s_wait_asynccnt 0x0" ::: "memory");
#endif
#else
  for (int i = tid; i < VEC4; i += NTHREADS) dstv[i] = src[i];
#endif
  __syncthreads();

  // ---- pass 1: row max (from LDS) ----
  float m = -3.402823466e38f;
  for (int i = tid; i < VEC4; i += NTHREADS) {
    float4 v = dstv[i];
    m = fmaxf(m, fmaxf(fmaxf(v.x, v.y), fmaxf(v.z, v.w)));
  }
  const float bm = block_max(m, red, tid);

  // ---- pass 2: sum of exp(x - max) ----
  float s = 0.0f;
  for (int i = tid; i < VEC4; i += NTHREADS) {
    float4 v = dstv[i];
    s += __expf(v.x - bm) + __expf(v.y - bm) + __expf(v.z - bm) + __expf(v.w - bm);
  }
  const float bs = block_sum(s, red, tid);
  const float logZ = bm + __logf(bs);

  // ---- pass 3: focal-weighted log-prob sum: sum_c (1-p)^3 * logp ----
  float f = 0.0f;
  for (int i = tid; i < VEC4; i += NTHREADS) {
    float4 v = dstv[i];
    {
      float ls = v.x - logZ; float om = 1.0f - __expf(ls); f += om * om * om * ls;
    }{
      float ls = v.y - logZ; float om = 1.0f - __expf(ls); f += om * om * om * ls;
    }{
      float ls = v.z - logZ; float om = 1.0f - __expf(ls); f += om * om * om * ls;
    }{
      float ls = v.w - logZ; float om = 1.0f - __expf(ls); f += om * om * om * ls;
    }
  }
  const float bf = block_sum(f, red, tid);

  if (tid == 0) {
    // smooth = onehot*(1-S) + S/C. Uniform S/C part is bf; target gets extra (1-S).
    float lst = row[tgt] - logZ;
    float omt = 1.0f - __expf(lst);
    float wt  = omt * omt * omt;
    float per_tok = -((SMOOTH / (float)CLASSES) * bf + COMPL * wt * lst);
    tok_loss[tok] = per_tok;
    tok_mask[tok] = 1.0f;
  }
}

// ---------- finalize: deterministic fixed-order reduction of 8192 tokens ----------

__global__ __launch_bounds__(NTHREADS) void focal_ls_ce_finalize(
    const float* __restrict__ tok_loss, const float* __restrict__ tok_mask,
    float* __restrict__ out, int n_tok) {
  __shared__ float redA[NWAVES];
  __shared__ float redB[NWAVES];
  const int tid = threadIdx.x;
  float s = 0.0f, c = 0.0f;
  for (int i = tid; i < n_tok; i += NTHREADS) { s += tok_loss[i]; c += tok_mask[i]; }
  float ws = wave_sum(s);
  float wc = wave_sum(c);
  if ((tid & 31) == 0) { redA[tid >> 5] = ws; redB[tid >> 5] = wc; }
  __syncthreads();
  if (tid == 0) {
    float S = redA[0], C = redB[0];
    #pragma unroll
    for (int i = 1; i < NWAVES; ++i) { S += redA[i]; C += redB[i]; }
    out[0] = S / C;
  }
}

extern "C" void kernel_launch(void* const* d_in, const int* in_sizes, int n_in,
                              void* d_out, int out_size, void* d_ws, size_t ws_size,
                              hipStream_t stream) {
  const float* logits = (const float*)d_in[0];
  const int*   target = (const int*)d_in[1];
  const int n_tok = in_sizes[1];                      // B*S = 8192

  float* tok_loss = (float*)d_ws;                     // n_tok floats
  float* tok_mask = tok_loss + n_tok;                 // n_tok floats (64KB total)

  focal_ls_ce_rows<<<n_tok, NTHREADS, 0, stream>>>(logits, target, tok_loss, tok_mask);
  focal_ls_ce_finalize<<<1, NTHREADS, 0, stream>>>(tok_loss, tok_mask, (float*)d_out, n_tok);
}